// GraphSAGEFraud_29446295781810
// MI455X (gfx1250) — compile-verified
//
#include <hip/hip_runtime.h>

#define N_NODES 50000
#define N_EDGES 800000
#define IN_CH   64
#define HIDDEN  16
#define OUT_CH  8

typedef __attribute__((ext_vector_type(2))) float v2f;
typedef __attribute__((ext_vector_type(8))) float v8f;

// ---------------------------------------------------------------- zero scratch
__global__ __launch_bounds__(256) void zero_ws(float* __restrict__ ws, int n) {
    int i = blockIdx.x * blockDim.x + threadIdx.x;
    if (i < n) ws[i] = 0.0f;
}

// ---------------------------------------------------------------- layer1 GEMMs
// One wave handles one 16-row tile of x.
//   p1 = x @ W1l^T            (projected features, aggregated later)
//   r1 = x @ W1r^T + b1       (root term; bias folded here, sum is associative)
// K=64 split into 16 chunks of 4 for V_WMMA_F32_16X16X4_F32 (exact fp32).
__global__ __launch_bounds__(32) void gemm1(const float* __restrict__ x,
                                            const float* __restrict__ W1l,
                                            const float* __restrict__ W1r,
                                            const float* __restrict__ b1,
                                            float* __restrict__ p1,
                                            float* __restrict__ r1) {
    const int lane = threadIdx.x;
    const int lo = lane & 15;      // M for A, N for B/C/D
    const int hi = lane >> 4;      // selects K pair (A/B) and M+8 half (C/D)
    const int rowBase = blockIdx.x * 16;

    v8f acc1 = {};   // x @ W1l^T
    v8f acc2 = {};   // x @ W1r^T

    const float* xrow = x   + (rowBase + lo) * IN_CH + 2 * hi;
    const float* wl   = W1l + lo * IN_CH + 2 * hi;   // B[k][n] = W1l[n][k]
    const float* wr   = W1r + lo * IN_CH + 2 * hi;

#pragma unroll
    for (int kc = 0; kc < 16; ++kc) {
        v2f a  = *(const v2f*)(xrow + kc * 4);
        v2f bl = *(const v2f*)(wl   + kc * 4);
        v2f br = *(const v2f*)(wr   + kc * 4);
        acc1 = __builtin_amdgcn_wmma_f32_16x16x4_f32(false, a, false, bl,
                                                     (short)0, acc1, false, false);
        acc2 = __builtin_amdgcn_wmma_f32_16x16x4_f32(false, a, false, br,
                                                     (short)0, acc2, false, false);
    }

    const float bias = b1[lo];
#pragma unroll
    for (int v = 0; v < 8; ++v) {
        const int row = rowBase + v + 8 * hi;      // C/D layout: M = v + 8*hi, N = lo
        p1[row * HIDDEN + lo] = acc1[v];
        r1[row * HIDDEN + lo] = acc2[v] + bias;
    }
}

// ---------------------------------------------------------------- layer2 GEMM
// B packs both weight matrices: cols 0..7 = W2l^T, cols 8..15 = W2r^T.
// One WMMA chain (K=16 -> 4 chunks) yields p2 (cols 0-7) and r2 (cols 8-15).
__global__ __launch_bounds__(32) void gemm2(const float* __restrict__ h,
                                            const float* __restrict__ W2l,
                                            const float* __restrict__ W2r,
                                            const float* __restrict__ b2,
                                            float* __restrict__ p2,
                                            float* __restrict__ r2) {
    const int lane = threadIdx.x;
    const int lo = lane & 15;
    const int hi = lane >> 4;
    const int rowBase = blockIdx.x * 16;

    v8f acc = {};
    const float* hrow = h + (rowBase + lo) * HIDDEN + 2 * hi;
    // Pointer select (no EXEC divergence before WMMA).
    const float* wp = (lo < 8) ? (W2l + lo * HIDDEN + 2 * hi)
                               : (W2r + (lo - 8) * HIDDEN + 2 * hi);
#pragma unroll
    for (int kc = 0; kc < 4; ++kc) {
        v2f a = *(const v2f*)(hrow + kc * 4);
        v2f b = *(const v2f*)(wp   + kc * 4);
        acc = __builtin_amdgcn_wmma_f32_16x16x4_f32(false, a, false, b,
                                                    (short)0, acc, false, false);
    }
#pragma unroll
    for (int v = 0; v < 8; ++v) {
        const int row = rowBase + v + 8 * hi;
        if (lo < 8) p2[row * OUT_CH + lo]       = acc[v];
        else        r2[row * OUT_CH + (lo - 8)] = acc[v] + b2[lo - 8];
    }
}

// ---------------------------------------------------------------- edge scatter
// Layer-1: 16 floats/edge into sum1 + degree count (deg shared by both layers).
__global__ __launch_bounds__(256) void scatter1(const long long* __restrict__ ei,
                                                const float* __restrict__ p1,
                                                float* __restrict__ sum1,
                                                float* __restrict__ deg) {
    int e = blockIdx.x * blockDim.x + threadIdx.x;
    if (e >= N_EDGES) return;
    const int s = (int)ei[e];
    const int d = (int)ei[N_EDGES + e];
    const float4* srcv = (const float4*)(p1 + (size_t)s * HIDDEN);
    float* dst = sum1 + (size_t)d * HIDDEN;
#pragma unroll
    for (int q = 0; q < 4; ++q) {
        float4 v = srcv[q];   // L2-resident gather (p1 = 3.2 MB << 192 MB L2)
        atomicAdd(dst + q * 4 + 0, v.x);
        atomicAdd(dst + q * 4 + 1, v.y);
        atomicAdd(dst + q * 4 + 2, v.z);
        atomicAdd(dst + q * 4 + 3, v.w);
    }
    atomicAdd(deg + d, 1.0f);
}

// Layer-2: 8 floats/edge into sum2.
__global__ __launch_bounds__(256) void scatter2(const long long* __restrict__ ei,
                                                const float* __restrict__ p2,
                                                float* __restrict__ sum2) {
    int e = blockIdx.x * blockDim.x + threadIdx.x;
    if (e >= N_EDGES) return;
    const int s = (int)ei[e];
    const int d = (int)ei[N_EDGES + e];
    const float4* srcv = (const float4*)(p2 + (size_t)s * OUT_CH);
    float* dst = sum2 + (size_t)d * OUT_CH;
#pragma unroll
    for (int q = 0; q < 2; ++q) {
        float4 v = srcv[q];
        atomicAdd(dst + q * 4 + 0, v.x);
        atomicAdd(dst + q * 4 + 1, v.y);
        atomicAdd(dst + q * 4 + 2, v.z);
        atomicAdd(dst + q * 4 + 3, v.w);
    }
}

// ---------------------------------------------------------------- combines
__global__ __launch_bounds__(256) void relu_combine(const float* __restrict__ sum1,
                                                    const float* __restrict__ r1,
                                                    const float* __restrict__ deg,
                                                    float* __restrict__ h) {
    int i = blockIdx.x * blockDim.x + threadIdx.x;
    if (i >= N_NODES * HIDDEN) return;
    const int node = i / HIDDEN;
    const float dg = fmaxf(deg[node], 1.0f);
    const float v = sum1[i] / dg + r1[i];
    h[i] = fmaxf(v, 0.0f);
}

__global__ __launch_bounds__(256) void out_combine(const float* __restrict__ sum2,
                                                   const float* __restrict__ r2,
                                                   const float* __restrict__ deg,
                                                   float* __restrict__ out) {
    int i = blockIdx.x * blockDim.x + threadIdx.x;
    if (i >= N_NODES * OUT_CH) return;
    const int node = i / OUT_CH;
    const float dg = fmaxf(deg[node], 1.0f);
    out[i] = sum2[i] / dg + r2[i];   // b2 already folded into r2
}

// ---------------------------------------------------------------- launch
extern "C" void kernel_launch(void* const* d_in, const int* in_sizes, int n_in,
                              void* d_out, int out_size, void* d_ws, size_t ws_size,
                              hipStream_t stream) {
    const float*     x   = (const float*)d_in[0];
    const float*     W1l = (const float*)d_in[1];
    const float*     W1r = (const float*)d_in[2];
    const float*     b1  = (const float*)d_in[3];
    const float*     W2l = (const float*)d_in[4];
    const float*     W2r = (const float*)d_in[5];
    const float*     b2  = (const float*)d_in[6];
    const long long* ei  = (const long long*)d_in[7];   // int64 edge_index [2, E]
    float* out = (float*)d_out;

    // Workspace layout (floats). deg+sum1+sum2 contiguous -> single zero pass.
    float* ws   = (float*)d_ws;
    float* deg  = ws;                          // 50000
    float* sum1 = deg  + N_NODES;              // 800000
    float* sum2 = sum1 + N_NODES * HIDDEN;     // 400000
    float* p1   = sum2 + N_NODES * OUT_CH;     // 800000
    float* r1   = p1   + N_NODES * HIDDEN;     // 800000
    float* h    = r1   + N_NODES * HIDDEN;     // 800000
    float* p2   = h    + N_NODES * HIDDEN;     // 400000
    float* r2   = p2   + N_NODES * OUT_CH;     // 400000
    // total ~4.45M floats (~17.8 MB)

    const int zn = N_NODES + N_NODES * HIDDEN + N_NODES * OUT_CH;
    zero_ws<<<(zn + 255) / 256, 256, 0, stream>>>(deg, zn);

    // Layer 1: project first (linearity of segment-mean), then aggregate.
    gemm1<<<N_NODES / 16, 32, 0, stream>>>(x, W1l, W1r, b1, p1, r1);
    scatter1<<<(N_EDGES + 255) / 256, 256, 0, stream>>>(ei, p1, sum1, deg);
    relu_combine<<<(N_NODES * HIDDEN + 255) / 256, 256, 0, stream>>>(sum1, r1, deg, h);

    // Layer 2.
    gemm2<<<N_NODES / 16, 32, 0, stream>>>(h, W2l, W2r, b2, p2, r2);
    scatter2<<<(N_EDGES + 255) / 256, 256, 0, stream>>>(ei, p2, sum2);
    out_combine<<<(N_NODES * OUT_CH + 255) / 256, 256, 0, stream>>>(sum2, r2, deg, out);
}